// MyModel_61933428412208
// MI455X (gfx1250) — compile-verified
//
#include <hip/hip_runtime.h>

// FNO spectral conv: out[b,o,s] = sum_i x[b,i,s] * w[i,o,s]  (complex, f32)
// B=128, I=16, O=24, S=24^3=13824. Per-s GEMM mapped onto V_WMMA_F32_16X16X4_F32.
// Staging: coalesced global -> swizzled LDS via GLOBAL_LOAD_ASYNC_TO_LDS_B32.

typedef __attribute__((ext_vector_type(2))) float v2f;
typedef __attribute__((ext_vector_type(8))) float v8f;

#define AS1 __attribute__((address_space(1)))
#define AS3 __attribute__((address_space(3)))

#define B_TOT 128
#define I_CH  16
#define O_CH  24
#define S_TOT 13824
#define S_BLK 16
#define B_TILE 16
#define NTHREADS 256

// LDS layout (word offsets). x tile: [ss][kp(8)][b(16)][2] + 2-word pad per ss.
// w tile: [ss][kp(8)][n(32, o padded)][2] + 2-word pad per ss.
// out staging: [bb(16)][o(24) stride 17][ss(16)].
#define XS_STRIDE 258
#define WS_STRIDE 514
#define O_OSTRIDE 17
#define O_BSTRIDE (24 * 17)
#define OFF_XR 0
#define OFF_XI (OFF_XR + S_BLK * XS_STRIDE)
#define OFF_WR (OFF_XI + S_BLK * XS_STRIDE)
#define OFF_WI (OFF_WR + S_BLK * WS_STRIDE)
#define OFF_OR (OFF_WI + S_BLK * WS_STRIDE)
#define OFF_OI (OFF_OR + B_TILE * O_BSTRIDE)
#define LDS_WORDS (OFF_OI + B_TILE * O_BSTRIDE)

// One 4-byte async copy: global (coalesced along s) -> LDS (swizzled), ASYNCcnt.
__device__ __forceinline__ void async_g2l_b32(const float* gptr, float* smem,
                                              int lds_word, int cpol_nt) {
#if __has_builtin(__builtin_amdgcn_global_load_async_to_lds_b32)
  if (cpol_nt)
    __builtin_amdgcn_global_load_async_to_lds_b32(
        (AS1 int*)gptr, (AS3 int*)&smem[lds_word], 0, 1 /* TH_NT */);
  else
    __builtin_amdgcn_global_load_async_to_lds_b32(
        (AS1 int*)gptr, (AS3 int*)&smem[lds_word], 0, 0);
#else
  // Fallback: GVS mode, dynamic-LDS base is offset 0 of the WG allocation.
  unsigned lds_off = (unsigned)lds_word * 4u;
  unsigned voff = 0;
  asm volatile("global_load_async_to_lds_b32 %0, %1, %2"
               :: "v"(lds_off), "v"(voff), "s"(gptr)
               : "memory");
#endif
}

__device__ __forceinline__ void wait_async_then_barrier() {
#if __has_builtin(__builtin_amdgcn_s_wait_asynccnt)
  __builtin_amdgcn_s_wait_asynccnt(0);
#else
  asm volatile("s_wait_asynccnt 0x0" ::: "memory");
#endif
  __syncthreads();
}

extern "C" __global__ __launch_bounds__(NTHREADS)
void fno_spectral_wmma(const float* __restrict__ xr, const float* __restrict__ xi,
                       const float* __restrict__ wr, const float* __restrict__ wi,
                       float* __restrict__ out) {
  extern __shared__ float smem[];
  const int tid = threadIdx.x;
  const int s0 = blockIdx.x * S_BLK;   // spatial block
  const int b0 = blockIdx.y * B_TILE;  // batch (GEMM-M) tile

  // ---- stage X tile (async, NT: 226MB stream, read once) ----
  for (int l = tid; l < B_TILE * I_CH * S_BLK; l += NTHREADS) {
    int ss = l & 15, i = (l >> 4) & 15, bb = l >> 8;
    size_t g = ((size_t)(b0 + bb) * I_CH + i) * S_TOT + s0 + ss;
    int d = ss * XS_STRIDE + (i >> 1) * 32 + bb * 2 + (i & 1);
    async_g2l_b32(&xr[g], smem, OFF_XR + d, 1);
    async_g2l_b32(&xi[g], smem, OFF_XI + d, 1);
  }
  // ---- stage W tile (async, RT: 43MB total lives in 192MB L2) ----
  for (int l = tid; l < I_CH * O_CH * S_BLK; l += NTHREADS) {
    int ss = l & 15, o = (l >> 4) % 24, i = l / (O_CH * S_BLK);
    size_t g = ((size_t)i * O_CH + o) * S_TOT + s0 + ss;
    int d = ss * WS_STRIDE + (i >> 1) * 64 + o * 2 + (i & 1);
    async_g2l_b32(&wr[g], smem, OFF_WR + d, 0);
    async_g2l_b32(&wi[g], smem, OFF_WI + d, 0);
  }
  // ---- zero-pad o = 24..31 so the second N-tile is clean ----
  for (int l = tid; l < 1024; l += NTHREADS) {
    int ss = l & 15, o = 24 + ((l >> 4) & 7), kp = l >> 7;
    int d = ss * WS_STRIDE + kp * 64 + o * 2;
    smem[OFF_WR + d] = 0.f; smem[OFF_WR + d + 1] = 0.f;
    smem[OFF_WI + d] = 0.f; smem[OFF_WI + d + 1] = 0.f;
  }
  wait_async_then_barrier();

  const int lane  = tid & 31;
  const int wv    = tid >> 5;   // 8 waves
  const int rowc  = lane & 15;  // A: M (batch) index / B: N (out-ch) index
  const int khalf = lane >> 4;  // K-pair select per WMMA f32 operand layout

  for (int j = 0; j < 2; ++j) {      // each wave owns 2 of the 16 s values
    const int ss = wv * 2 + j;
    // A operands (both complex parts) for all 4 K-steps: one b64 each.
    v2f ar[4], ai[4];
#pragma unroll
    for (int k = 0; k < 4; ++k) {
      int kp = k * 2 + khalf;
      int d = ss * XS_STRIDE + kp * 32 + rowc * 2;
      ar[k] = *(const v2f*)&smem[OFF_XR + d];
      ai[k] = *(const v2f*)&smem[OFF_XI + d];
    }
#pragma unroll
    for (int nt = 0; nt < 2; ++nt) {
      const int n = nt * 16 + rowc;
      v8f accr = {0.f, 0.f, 0.f, 0.f, 0.f, 0.f, 0.f, 0.f};
      v8f acci = {0.f, 0.f, 0.f, 0.f, 0.f, 0.f, 0.f, 0.f};
#pragma unroll
      for (int k = 0; k < 4; ++k) {
        int kp = k * 2 + khalf;
        int d = ss * WS_STRIDE + kp * 64 + n * 2;
        v2f br = *(const v2f*)&smem[OFF_WR + d];
        v2f bi = *(const v2f*)&smem[OFF_WI + d];
        v2f nbi = -bi;  // f32 WMMA has no A/B NEG modifier (CNeg only)
        accr = __builtin_amdgcn_wmma_f32_16x16x4_f32(false, ar[k], false, br,
                                                     (short)0, accr, false, false);
        accr = __builtin_amdgcn_wmma_f32_16x16x4_f32(false, ai[k], false, nbi,
                                                     (short)0, accr, false, false);
        acci = __builtin_amdgcn_wmma_f32_16x16x4_f32(false, ar[k], false, bi,
                                                     (short)0, acci, false, false);
        acci = __builtin_amdgcn_wmma_f32_16x16x4_f32(false, ai[k], false, br,
                                                     (short)0, acci, false, false);
      }
      // C layout: VGPR r, lanes 0-15 -> M=r, lanes 16-31 -> M=r+8; N = lane%16.
      const int o = nt * 16 + rowc;
      if (o < O_CH) {
#pragma unroll
        for (int r = 0; r < 8; ++r) {
          int bb = r + 8 * khalf;
          int d = bb * O_BSTRIDE + o * O_OSTRIDE + ss;
          smem[OFF_OR + d] = accr[r];
          smem[OFF_OI + d] = acci[r];
        }
      }
    }
  }
  __syncthreads();

  // ---- write out coalesced along s (NT: 340MB stream, keep w in L2) ----
  const size_t BOS = (size_t)B_TOT * O_CH * S_TOT;
  for (int l = tid; l < B_TILE * O_CH * S_BLK; l += NTHREADS) {
    int ss = l & 15, o = (l >> 4) % 24, bb = l / (O_CH * S_BLK);
    size_t g = ((size_t)(b0 + bb) * O_CH + o) * S_TOT + s0 + ss;
    int d = bb * O_BSTRIDE + o * O_OSTRIDE + ss;
    __builtin_nontemporal_store(smem[OFF_OR + d], &out[g]);
    __builtin_nontemporal_store(smem[OFF_OI + d], &out[BOS + g]);
  }
}

extern "C" void kernel_launch(void* const* d_in, const int* in_sizes, int n_in,
                              void* d_out, int out_size, void* d_ws, size_t ws_size,
                              hipStream_t stream) {
  (void)in_sizes; (void)n_in; (void)out_size; (void)d_ws; (void)ws_size;
  const float* xr = (const float*)d_in[0];
  const float* xi = (const float*)d_in[1];
  const float* wr = (const float*)d_in[2];
  const float* wi = (const float*)d_in[3];
  float* out = (float*)d_out;

  dim3 grid(S_TOT / S_BLK, B_TOT / B_TILE);  // 864 x 8 workgroups
  dim3 block(NTHREADS);                      // 8 waves (wave32)
  size_t shmem = (size_t)LDS_WORDS * sizeof(float);  // ~148 KB dynamic LDS
  hipLaunchKernelGGL(fno_spectral_wmma, grid, block, shmem, stream,
                     xr, xi, wr, wi, out);
}